// DeformableConvBlock_15848429322703
// MI455X (gfx1250) — compile-verified
//
#include <hip/hip_runtime.h>

typedef __attribute__((ext_vector_type(2))) float v2f;
typedef __attribute__((ext_vector_type(8))) float v8f;
typedef __attribute__((ext_vector_type(4))) unsigned int v4u;
typedef __attribute__((ext_vector_type(4))) int v4i;
typedef __attribute__((ext_vector_type(8))) int v8i;

#define BATCH 4
#define CIN   64
#define COUT  64
#define Hdim  128
#define Wdim  128
#define KKTAP 9
#define HW    (Hdim * Wdim)        // 16384
#define RED   (CIN * KKTAP)        // 576
#define NOFF  (2 * KKTAP)          // 18
#define SLICE (CIN * COUT)         // 4096 floats per tap
#define LROW  160                  // LDS row stride in floats (128 data + 32 pad)
#define LSLICE ((SLICE / 128) * LROW) // 5120 floats per LDS slice (20 KB)

// ---------------------------------------------------------------------------
// Kernel 1: weight (Cout, Cin, 3, 3) -> pair-interleaved
//   Wt2[kk*4096 + (c/2)*128 + o*2 + (c&1)]
// so a lane's WMMA A-fragment (K-pair {c, c+1} for output o) is one
// contiguous 8-byte read after the TDM stages it into LDS.
// ---------------------------------------------------------------------------
__global__ void __launch_bounds__(256) k_transpose_w(const float* __restrict__ w,
                                                     float* __restrict__ wt2) {
    int id = blockIdx.x * 256 + threadIdx.x;      // 36864 elements
    int o   = id / RED;
    int rem = id % RED;
    int c   = rem / KKTAP;
    int kk  = rem % KKTAP;
    wt2[kk * SLICE + (c >> 1) * (COUT * 2) + o * 2 + (c & 1)] = w[id];
}

// ---------------------------------------------------------------------------
// Kernel 2: offset conv (standard 3x3, pad 1): off[b, ch, h, w], ch in [0,18)
// ch is uniform per 256-thread block (HW is a multiple of 256), so we stage
// that channel's 576 weights in LDS.
// ---------------------------------------------------------------------------
__global__ void __launch_bounds__(256) k_offset_conv(const float* __restrict__ x,
                                                     const float* __restrict__ ow,
                                                     const float* __restrict__ ob,
                                                     float* __restrict__ off) {
    __shared__ float sw[RED];
    int id  = blockIdx.x * 256 + threadIdx.x;
    int pix = id & (HW - 1);
    int bc  = id >> 14;                 // / HW
    int ch  = bc % NOFF;
    int b   = bc / NOFF;
    int h   = pix >> 7;
    int w   = pix & (Wdim - 1);

    for (int t = threadIdx.x; t < RED; t += 256) sw[t] = ow[ch * RED + t];
    __syncthreads();

    float acc = ob[ch];
    const float* xb = x + (size_t)b * CIN * HW;
    for (int ci = 0; ci < CIN; ++ci) {
        const float* xc  = xb + (size_t)ci * HW;
        const float* swc = sw + ci * KKTAP;
#pragma unroll
        for (int i = 0; i < 3; ++i) {
            int yy = h + i - 1;
            bool yok = (yy >= 0) & (yy < Hdim);
#pragma unroll
            for (int j = 0; j < 3; ++j) {
                int xx = w + j - 1;
                bool ok = yok & (xx >= 0) & (xx < Wdim);
                float xv = ok ? xc[yy * Wdim + xx] : 0.0f;
                acc = fmaf(xv, swc[i * 3 + j], acc);
            }
        }
    }
    off[id] = acc;
}

// ---------------------------------------------------------------------------
// TDM: async DMA one 16 KB weight slice (4096 f32) global -> LDS, inserting a
// 32-DWORD pad every 128 DWORDs (LDS row stride 160 ≡ 32 mod 64 banks, so the
// two half-wave K-pair groups read disjoint bank ranges).
// ---------------------------------------------------------------------------
__device__ __forceinline__ void tdm_load_slice(const float* gsrc, unsigned int lds_byte_off) {
    unsigned long long ga = (unsigned long long)(uintptr_t)gsrc;
    const unsigned int n = SLICE;                 // 4096 elements, 1 row
    v4u g0 = { 1u,                                // count=1, not restore, no gather
               lds_byte_off,                      // lds_addr (bytes)
               (unsigned int)ga,                  // global_addr[31:0]
               (unsigned int)(ga >> 32) | (2u << 30) };  // addr[56:32] | type=2
    v8i g1 = { (int)((2u << 16)                   // data_size=2 (4 bytes)
                     | (1u << 20)                 // pad_enable
                     | (6u << 22)                 // pad_interval: every 128 DWORDs
                     | (31u << 25)),              // pad_amount: 32 DWORDs
               (int)((n & 0xFFFFu) << 16),        // tensor_dim0[15:0] @ bits 63:48
               (int)(((n >> 16) & 0xFFFFu) | (1u << 16)), // dim0 hi | tensor_dim1=1
               (int)((n & 0xFFFFu) << 16),        // tensor_dim1 hi=0 | tile_dim0=n
               (int)1,                            // tile_dim1=1, tile_dim2=0
               (int)n,                            // tensor_dim0_stride[31:0]
               0, 0 };                            // stride hi, tensor_dim1_stride
    v4i gz4 = { 0, 0, 0, 0 };
    v8i gz8 = { 0, 0, 0, 0, 0, 0, 0, 0 };
    __builtin_amdgcn_tensor_load_to_lds(g0, g1, gz4, gz4, gz8, 0);
}

// ---------------------------------------------------------------------------
// Kernel 3: main deformable contraction via V_WMMA_F32_16X16X4_F32.
// One wave -> 64 output channels x 16 pixels (b, h, w0..w0+15).
// K = 576 walked as 9 taps x 16 channel-quads. Per tap, wave 0 TDM-stages the
// next tap's 16 KB weight slice into the other LDS buffer while all 8 waves
// compute from the current one (weights via ds_load_b64, gathers via VMEM).
// ---------------------------------------------------------------------------
__global__ void __launch_bounds__(256) k_deform_main(const float* __restrict__ x,
                                                     const float* __restrict__ off,
                                                     const float* __restrict__ wt2,
                                                     const float* __restrict__ bias,
                                                     float* __restrict__ out) {
    __shared__ float swt[2 * LSLICE];             // 40 KB double buffer

    const int lane = threadIdx.x & 31;
    const int wave = threadIdx.x >> 5;
    const int wid  = blockIdx.x * 8 + wave;       // 4096 waves total
    const int b    = wid >> 10;                   // / (128 * 8)
    const int rem  = wid & 1023;
    const int h    = rem >> 3;
    const int w0   = (rem & 7) << 4;
    const int p    = lane & 15;                   // pixel within tile / output lane
    const int half = lane >> 4;                   // K-pair selector for A/B frags
    const int w    = w0 + p;

    const float* xb   = x + (size_t)b * CIN * HW;
    const float* offb = off + (size_t)b * NOFF * HW;

    v8f acc[4];
#pragma unroll
    for (int n = 0; n < 4; ++n) acc[n] = (v8f){};

    if (wave == 0) tdm_load_slice(wt2, 0u);       // prefetch tap 0 -> buffer 0

#pragma unroll 1
    for (int kk = 0; kk < KKTAP; ++kk) {
        if (wave == 0) __builtin_amdgcn_s_wait_tensorcnt(0);
        __syncthreads();                          // slice kk visible; prev buffer free

        // The TDM builtin carries the LDS destination only as a descriptor
        // value, so alias analysis cannot see that swt is written. Force the
        // compiler to treat the staged buffer as modified memory here — this
        // keeps the ds_load_b64 A-fragment reads alive and ordered.
        asm volatile("" : : "v"(&swt[0]) : "memory");

        if (wave == 0 && kk + 1 < KKTAP)
            tdm_load_slice(wt2 + (size_t)(kk + 1) * SLICE,
                           (unsigned int)(((kk + 1) & 1) * LSLICE * sizeof(float)));

        const float* lwt = swt + (kk & 1) * LSLICE;

        // per-pixel deformed sampling position for this tap
        float dy = offb[(2 * kk + 0) * HW + h * Wdim + w];
        float dx = offb[(2 * kk + 1) * HW + h * Wdim + w];
        float py = dy + (float)(h + kk / 3 - 1);
        float px = dx + (float)(w + kk % 3 - 1);
        float y0f = floorf(py), x0f = floorf(px);
        float ly = py - y0f, lx = px - x0f;
        int y0 = (int)y0f, x0 = (int)x0f;
        int y1 = y0 + 1,  x1 = x0 + 1;
        float m00 = (y0 >= 0 && y0 < Hdim && x0 >= 0 && x0 < Wdim) ? 1.f : 0.f;
        float m01 = (y0 >= 0 && y0 < Hdim && x1 >= 0 && x1 < Wdim) ? 1.f : 0.f;
        float m10 = (y1 >= 0 && y1 < Hdim && x0 >= 0 && x0 < Wdim) ? 1.f : 0.f;
        float m11 = (y1 >= 0 && y1 < Hdim && x1 >= 0 && x1 < Wdim) ? 1.f : 0.f;
        int y0c = min(max(y0, 0), Hdim - 1), y1c = min(max(y1, 0), Hdim - 1);
        int x0c = min(max(x0, 0), Wdim - 1), x1c = min(max(x1, 0), Wdim - 1);
        int l00 = y0c * Wdim + x0c, l01 = y0c * Wdim + x1c;
        int l10 = y1c * Wdim + x0c, l11 = y1c * Wdim + x1c;
        float w00 = (1.f - ly) * (1.f - lx) * m00;
        float w01 = (1.f - ly) * lx * m01;
        float w10 = ly * (1.f - lx) * m10;
        float w11 = ly * lx * m11;

#pragma unroll 4
        for (int c0 = 0; c0 < CIN; c0 += 4) {
            // B-fragment: sampled values, 2 channels per lane (K = c0+2*half, +1)
            const float* pc = xb + (size_t)(c0 + 2 * half) * HW;
            float a0 = w00 * pc[l00] + w01 * pc[l01] + w10 * pc[l10] + w11 * pc[l11];
            pc += HW;
            float a1 = w00 * pc[l00] + w01 * pc[l01] + w10 * pc[l10] + w11 * pc[l11];
            v2f bfrag; bfrag.x = a0; bfrag.y = a1;

            // A-fragments from LDS: pair-row (c0/2 + half), 8B per lane
            const float* lrow = lwt + ((c0 >> 1) + half) * LROW + p * 2;
#pragma unroll
            for (int n = 0; n < 4; ++n) {
                v2f afrag = *(const v2f*)(lrow + n * 32);
                acc[n] = __builtin_amdgcn_wmma_f32_16x16x4_f32(
                    false, afrag, false, bfrag, (short)0, acc[n], false, false);
            }
        }
    }

    // D layout: VGPR j holds M = j + 8*half (output channel - n*16), N = p (pixel)
    const int base = ((b * COUT) * Hdim + h) * Wdim + w0 + p;
#pragma unroll
    for (int n = 0; n < 4; ++n) {
#pragma unroll
        for (int j = 0; j < 8; ++j) {
            int o = n * 16 + 8 * half + j;
            out[base + o * HW] = acc[n][j] + bias[o];
        }
    }
}

// ---------------------------------------------------------------------------
extern "C" void kernel_launch(void* const* d_in, const int* in_sizes, int n_in,
                              void* d_out, int out_size, void* d_ws, size_t ws_size,
                              hipStream_t stream) {
    const float* x        = (const float*)d_in[0];
    const float* offset_w = (const float*)d_in[1];
    const float* offset_b = (const float*)d_in[2];
    const float* weight   = (const float*)d_in[3];
    const float* bias     = (const float*)d_in[4];
    float* out = (float*)d_out;

    // workspace layout: [Wt2: 9*4096 floats][offsets: B*18*H*W floats]
    float* wt2  = (float*)d_ws;
    float* offs = (float*)((char*)d_ws + (size_t)KKTAP * SLICE * sizeof(float));

    k_transpose_w<<<(COUT * RED) / 256, 256, 0, stream>>>(weight, wt2);
    k_offset_conv<<<(BATCH * NOFF * HW) / 256, 256, 0, stream>>>(x, offset_w, offset_b, offs);
    k_deform_main<<<(BATCH * Hdim * (Wdim / 16)) / 8, 256, 0, stream>>>(x, offs, wt2, bias, out);
}